// hiddenLayer_76630806495591
// MI455X (gfx1250) — compile-verified
//
#include <hip/hip_runtime.h>

// CDNA5 / gfx1250: wave32, WMMA f32 16x16x4, TDM / async global->LDS weight stream
typedef float    v2f __attribute__((ext_vector_type(2)));
typedef float    v4f __attribute__((ext_vector_type(4)));
typedef float    v8f __attribute__((ext_vector_type(8)));
typedef unsigned v4u __attribute__((ext_vector_type(4)));
typedef int      v4i __attribute__((ext_vector_type(4)));
typedef int      v8i __attribute__((ext_vector_type(8)));

static constexpr int Nn = 8192;      // hidden neurons (output cols)
static constexpr int Kk = 8192;      // prev-layer neurons (reduction dim)
static constexpr int Bb = 8;         // batch (output rows)
static constexpr int KC = 16;        // K-split factor (partials)
static constexpr int KL = Kk / KC;   // 512 K per chunk
static constexpr int KT = 32;        // K per staged weight tile
static constexpr int WST = 136;      // LDS weight row stride (128+8: 2*136%64==16 -> no half-wave bank clash)
static constexpr int SST = KL + 4;   // LDS spike row stride

#if __has_builtin(__builtin_amdgcn_tensor_load_to_lds)
#define USE_TDM 1
#if __has_include(<hip/amd_detail/amd_gfx1250_TDM.h>)
#define TDM_6ARG 1
#endif
#else
#define USE_TDM 0
#endif

__device__ __forceinline__ void wait_async0() {
#if __has_builtin(__builtin_amdgcn_s_wait_asynccnt)
  __builtin_amdgcn_s_wait_asynccnt(0);
#else
  asm volatile("s_wait_asynccnt 0x0" ::: "memory");
#endif
}

__device__ __forceinline__ void wait_tensor0() {
#if __has_builtin(__builtin_amdgcn_s_wait_tensorcnt)
  __builtin_amdgcn_s_wait_tensorcnt(0);
#else
  asm volatile("s_wait_tensorcnt 0x0" ::: "memory");
#endif
}

#if USE_TDM
// One TDM op DMAs a KT x 128 f32 tile (16 KB) global->LDS, padding each
// 128-DWORD row with 8 DWORDs so LDS rows land on the WST=136 stride.
__device__ __forceinline__ void tdm_issue_tile(unsigned long long ga, unsigned lds_off) {
  v4u g0;
  g0[0] = 1u;                                            // count=1, user descriptor
  g0[1] = lds_off;                                       // lds_addr (bytes)
  g0[2] = (unsigned)(ga & 0xffffffffu);                  // global_addr[31:0]
  g0[3] = (unsigned)((ga >> 32) & 0x01ffffffu)           // global_addr[56:32]
          | 0x80000000u;                                 // type=2 ("image")
  v8i g1;
  g1[0] = (int)((2u << 16)        // data_size = 4 B
          | (1u << 20)            // pad_enable
          | (6u << 22)            // pad_interval: 128 DWORDs (one 512 B row)
          | (7u << 25));          // pad_amount: 8 DWORDs (32 B)
  g1[1] = (int)(8192u << 16);     // tensor_dim0 = 8192 (cols), low16
  g1[2] = (int)(8192u << 16);     // dim0 hi=0 | tensor_dim1 = 8192 (rows), low16
  g1[3] = (int)(128u << 16);      // dim1 hi=0 | tile_dim0 = 128
  g1[4] = 32;                     // tile_dim1 = 32, tile_dim2 = 0
  g1[5] = 8192;                   // tensor_dim0_stride = 8192 elements
  g1[6] = 0;
  g1[7] = 0;
  v4i gz = {0, 0, 0, 0};          // 2-D tile: groups 2/3 unused
#if defined(TDM_6ARG)
  v8i z8 = {0, 0, 0, 0, 0, 0, 0, 0};
  __builtin_amdgcn_tensor_load_to_lds(g0, g1, gz, gz, z8, 0);
#else
  __builtin_amdgcn_tensor_load_to_lds(g0, g1, gz, gz, 0);
#endif
}
#else
// Fallback: per-lane async copies, 4 x b128 per thread = one 16 KB tile/block.
__device__ __forceinline__ void async_issue4(const float* g0, const float* g1,
                                             const float* g2, const float* g3,
                                             unsigned l) {
  asm volatile("global_load_async_to_lds_b128 %0, %4, off th:TH_LOAD_NT\n\t"
               "global_load_async_to_lds_b128 %1, %5, off th:TH_LOAD_NT\n\t"
               "global_load_async_to_lds_b128 %2, %6, off th:TH_LOAD_NT\n\t"
               "global_load_async_to_lds_b128 %3, %7, off th:TH_LOAD_NT"
               :: "v"(l), "v"(l + 8u * WST * 4u), "v"(l + 16u * WST * 4u),
                  "v"(l + 24u * WST * 4u),
                  "v"(g0), "v"(g1), "v"(g2), "v"(g3)
               : "memory");
}
#endif

__launch_bounds__(256, 1)
__global__ void snn_gemm(const float* __restrict__ spikes,
                         const float* __restrict__ weights,
                         float* __restrict__ dst) {
  __shared__ __align__(16) float lds_w[2 * KT * WST];  // 34816 B, double-buffered
  __shared__ __align__(16) float lds_s[Bb * SST];      // 16512 B
  __shared__ __align__(16) float lds_z[KL];            //  2048 B, stays zero (A pad rows)

  const int tid  = threadIdx.x;
  const int wave = tid >> 5;
  const int lane = tid & 31;
  const int wl   = lane & 15;
  const int hi   = lane >> 4;
  const int n0   = blockIdx.x * 128;
  const int nloc = wave * 16 + wl;

  const float* abase = (wl < Bb) ? &lds_s[(wl & 7) * SST] : &lds_z[0];
  for (int i = tid; i < KL; i += 256) lds_z[i] = 0.0f;

  const unsigned lw0      = (unsigned)(uintptr_t)&lds_w[0];  // LDS byte offset, buffer 0
  const unsigned lbufstep = (unsigned)(KT * WST * 4);        // 17408 B

#if !USE_TDM
  const int r0 = tid >> 5;
  const int c0 = (tid & 31) << 2;
  const unsigned ldst = lw0 + (unsigned)(r0 * WST + c0) * 4u;
#endif

  v8f acc = {};

  for (int ch = blockIdx.y; ch < KC; ch += gridDim.y) {
    const int k0 = ch * KL;
    __syncthreads();

    // Stage spikes chunk [8][KL] -> LDS
    for (int i = tid; i < (Bb * KL) / 4; i += 256) {
      const int r = i >> 7;
      const int c = (i & 127) << 2;
      v4f v = *(const v4f*)(spikes + (size_t)r * Kk + k0 + c);
      *(v4f*)&lds_s[r * SST + c] = v;
    }

    // --- prologue: tile 0 into buffer 0 ---
#if USE_TDM
    unsigned long long ga = (unsigned long long)(uintptr_t)weights
                          + ((unsigned long long)k0 * Nn + (unsigned)n0) * 4ull;
    if (wave == 0) { tdm_issue_tile(ga, lw0); wait_tensor0(); }
    ga += (unsigned long long)KT * Nn * 4ull;   // advance one tile of rows
#else
    const float* g0 = weights + (size_t)(k0 + r0) * Nn + n0 + c0;
    const float* g1 = g0 + (size_t)8  * Nn;
    const float* g2 = g0 + (size_t)16 * Nn;
    const float* g3 = g0 + (size_t)24 * Nn;
    async_issue4(g0, g1, g2, g3, ldst);
    g0 += (size_t)KT * Nn; g1 += (size_t)KT * Nn;
    g2 += (size_t)KT * Nn; g3 += (size_t)KT * Nn;
    wait_async0();
#endif
    __syncthreads();

    int cur = 0;
    for (int kk = 0; kk < KL; kk += KT) {
      if (kk + KT < KL) {   // prefetch next tile into the other buffer
#if USE_TDM
        if (wave == 0) tdm_issue_tile(ga, lw0 + (unsigned)(cur ^ 1) * lbufstep);
        ga += (unsigned long long)KT * Nn * 4ull;
#else
        async_issue4(g0, g1, g2, g3, ldst + (unsigned)(cur ^ 1) * lbufstep);
        g0 += (size_t)KT * Nn; g1 += (size_t)KT * Nn;
        g2 += (size_t)KT * Nn; g3 += (size_t)KT * Nn;
#endif
      }

      const float* wt = lds_w + cur * (KT * WST);
#pragma unroll
      for (int k2 = 0; k2 < KT; k2 += 4) {
        const int kb = k2 + 2 * hi;
        v2f b;                                   // B 4x16: rows {kb, kb+1} at column nloc
        b.x = wt[(kb + 0) * WST + nloc];
        b.y = wt[(kb + 1) * WST + nloc];
        v2f a = *(const v2f*)(abase + kk + kb);  // A 16x4: row m=wl (zero region if m>=8)
        acc = __builtin_amdgcn_wmma_f32_16x16x4_f32(
            false, a, false, b, (short)0, acc, false, false);
      }

#if USE_TDM
      if (wave == 0) wait_tensor0();
#else
      wait_async0();
#endif
      __syncthreads();
      cur ^= 1;
    }
  }

  if (lane < 16) {
    float* o = dst + (size_t)blockIdx.y * Bb * Nn + n0 + nloc;
#pragma unroll
    for (int r = 0; r < Bb; ++r) o[(size_t)r * Nn] = acc[r];
  }
}

__global__ void snn_reduce(const float* __restrict__ part, float* __restrict__ out) {
  const int g = blockIdx.x * blockDim.x + threadIdx.x;
  float s = 0.0f;
#pragma unroll
  for (int c = 0; c < KC; ++c) s += part[(size_t)c * Bb * Nn + g];
  out[g] = s;
}

extern "C" void kernel_launch(void* const* d_in, const int* in_sizes, int n_in,
                              void* d_out, int out_size, void* d_ws, size_t ws_size,
                              hipStream_t stream) {
  const float* spikes  = (const float*)d_in[0];   // [8, 8192] f32
  const float* weights = (const float*)d_in[1];   // [8192, 8192] f32
  float* out = (float*)d_out;                     // [8, 8192] f32

  const size_t need = (size_t)KC * Bb * Nn * sizeof(float);  // 4 MiB of partials
  if (ws_size >= need) {
    dim3 grid(Nn / 128, KC);    // 64 x 16 blocks, 8 waves each -> 8192 waves
    snn_gemm<<<grid, 256, 0, stream>>>(spikes, weights, (float*)d_ws);
    snn_reduce<<<(Bb * Nn) / 256, 256, 0, stream>>>((const float*)d_ws, out);
  } else {
    dim3 grid(Nn / 128, 1);     // fallback: accumulate all K in-register
    snn_gemm<<<grid, 256, 0, stream>>>(spikes, weights, out);
  }
}